// TransformerDecoder_51539607609
// MI455X (gfx1250) — compile-verified
//
#include <hip/hip_runtime.h>
#include <hip/hip_bf16.h>
#include <math.h>

// CDNA5 wave32 WMMA: D(16x16,f32) = A(16x4,f32) * B(4x16,f32) + C
typedef __attribute__((ext_vector_type(2))) float v2f;
typedef __attribute__((ext_vector_type(8))) float v8f;

#define ASTR 1032   // padded LDS row stride (1024 + 8) to break bank alignment

__device__ __forceinline__ v8f wmma4(v2f a, v2f b, v8f c) {
  // (neg_a, A, neg_b, B, c_mod, C, reuse_a, reuse_b)
  return __builtin_amdgcn_wmma_f32_16x16x4_f32(false, a, false, b, (short)0, c, false, false);
}
__device__ __forceinline__ v2f nt_v2(const float* p) {
  return __builtin_nontemporal_load((const v2f*)p);
}
__device__ __forceinline__ float nt_f(const float* p) {
  return __builtin_nontemporal_load(p);
}

// ---------------- Kernel 1: 8-head MHA, one block per batch ----------------
__global__ void __launch_bounds__(256)
mha1_kernel(const float* __restrict__ Q, const float* __restrict__ K,
            const float* __restrict__ V, const unsigned char* __restrict__ mask,
            float* __restrict__ attn_out)
{
  __shared__ float qs[512];
  __shared__ float attnL[8 * ASTR];

  const int b = blockIdx.x;
  const int tid = threadIdx.x;
  const int lane = tid & 31;
  const int wave = tid >> 5;
  const int lrow = lane & 15;   // row/col within fragment half
  const int khalf = lane >> 4;  // which k-pair (0: k0,k1  1: k2,k3)

  const float* Kb = K + (size_t)b * (1024 * 512);
  const float* Vb = V + (size_t)b * (1024 * 512);
  const unsigned char* mb = mask + (size_t)b * 1024;

  for (int i = tid; i < 128; i += 256)
    ((float4*)qs)[i] = ((const float4*)(Q + (size_t)b * 512))[i];
  __syncthreads();

  // ---- logits[h][n] = (K @ Qblk)/8, clip, mask.  64 key-tiles / 8 waves. ----
  for (int t = 0; t < 8; ++t) {
    const int m0 = (t * 8 + wave) << 4;
    v8f acc = {};
    const float* arow = Kb + (size_t)(m0 + lrow) * 512 + 2 * khalf;
    for (int kk = 0; kk < 512; kk += 4) {
      v2f a = nt_v2(arow + kk);                       // A: 16 keys x 4 k
      v2f bq = *(const v2f*)(qs + kk + 2 * khalf);    // B: only head col kk>>6 nonzero
      v2f bf = (lrow == (kk >> 6)) ? bq : (v2f)(0.f);
      acc = wmma4(a, bf, acc);
    }
    if (lrow < 8) {   // N = lrow = head
      union { v8f v; float f[8]; } u; u.v = acc;
      #pragma unroll
      for (int r = 0; r < 8; ++r) {
        const int n = m0 + r + 8 * khalf;             // M = key index
        float x = 10.f * tanhf(u.f[r] * 0.125f);      // /sqrt(64), clip
        if (mb[n]) x = -3.4028235e38f;
        attnL[lrow * ASTR + n] = x;
      }
    }
  }
  __syncthreads();

  // ---- per-head softmax: wave w owns head w (1024 elems over 32 lanes) ----
  {
    float* row = attnL + wave * ASTR;
    float m = -3.4028235e38f;
    for (int n = lane; n < 1024; n += 32) m = fmaxf(m, row[n]);
    #pragma unroll
    for (int o = 16; o > 0; o >>= 1) m = fmaxf(m, __shfl_xor(m, o, 32));
    float s = 0.f;
    for (int n = lane; n < 1024; n += 32) s += expf(row[n] - m);
    #pragma unroll
    for (int o = 16; o > 0; o >>= 1) s += __shfl_xor(s, o, 32);
    const float inv = 1.f / s;
    for (int n = lane; n < 1024; n += 32) row[n] = expf(row[n] - m) * inv;
  }
  __syncthreads();

  // ---- out = attnP(16x1024) @ V(1024x512): 32 col-tiles / 8 waves ----
  for (int t = 0; t < 4; ++t) {
    const int n0 = (t * 8 + wave) << 4;
    v8f acc = {};
    const float* arow = attnL + (lrow & 7) * ASTR + 2 * khalf;
    const float* vcol = Vb + n0 + lrow + (size_t)(2 * khalf) * 512;
    for (int kk = 0; kk < 1024; kk += 4) {
      v2f a = *(const v2f*)(arow + kk);
      a = (lrow < 8) ? a : (v2f)(0.f);                // pad heads 8..15 with zeros
      const float* vp = vcol + (size_t)kk * 512;
      v2f bf; bf.x = nt_f(vp); bf.y = nt_f(vp + 512); // B: 4 V-rows x 16 cols
      acc = wmma4(a, bf, acc);
    }
    // need D row == head(n0) (uniform per tile); held by lanes 0..15, vgpr r==head
    const int h = n0 >> 6;
    union { v8f v; float f[8]; } u; u.v = acc;
    float val = u.f[0];
    #pragma unroll
    for (int r = 1; r < 8; ++r) if (h == r) val = u.f[r];
    if (lane < 16) attn_out[(size_t)b * 512 + n0 + lane] = val;
  }
}

// ---------------- Kernel 2: dense GEMM  out = f(A @ W^T + bias [+ resid]) ----
__global__ void __launch_bounds__(256)
ffn_kernel(const float* __restrict__ A, const float* __restrict__ W,
           const float* __restrict__ bias, const float* __restrict__ resid,
           float* __restrict__ out, int apply_gelu)
{
  const int tid = threadIdx.x, lane = tid & 31, wave = tid >> 5;
  const int lrow = lane & 15, khalf = lane >> 4;
  const int wid = blockIdx.x * 8 + wave;   // 512 tiles = (256/16) x (512/16)
  const int m0 = (wid >> 5) << 4;
  const int n0 = (wid & 31) << 4;

  v8f acc = {};
  const float* ap = A + (size_t)(m0 + lrow) * 512 + 2 * khalf;
  const float* wp = W + (size_t)(n0 + lrow) * 512 + 2 * khalf;  // B^T[k][n] = W[n][k]
  for (int kk = 0; kk < 512; kk += 4) {
    v2f a = *(const v2f*)(ap + kk);
    v2f bf = *(const v2f*)(wp + kk);
    acc = wmma4(a, bf, acc);
  }
  union { v8f v; float f[8]; } u; u.v = acc;
  const int col = n0 + lrow;
  const float bv = bias[col];
  #pragma unroll
  for (int r = 0; r < 8; ++r) {
    const int row = m0 + r + 8 * khalf;
    float x = u.f[r] + bv;
    if (apply_gelu) x = 0.5f * x * (1.f + erff(x * 0.70710678118654752f)); // exact GELU
    else            x += resid[(size_t)row * 512 + col];
    out[(size_t)row * 512 + col] = x;
  }
}

// ---------------- Kernel 3: BatchNorm over batch axis (one thread / channel) ----
__global__ void __launch_bounds__(256)
bn_kernel(const float* __restrict__ X, const float* __restrict__ g,
          const float* __restrict__ bta, float* __restrict__ out)
{
  const int c = blockIdx.x * 256 + threadIdx.x;   // 512 channels
  float s = 0.f, s2 = 0.f;
  for (int r = 0; r < 256; ++r) {
    const float v = X[(size_t)r * 512 + c];
    s += v; s2 += v * v;
  }
  const float mu = s * (1.f / 256.f);
  const float var = s2 * (1.f / 256.f) - mu * mu;
  const float sc = rsqrtf(var + 1e-5f) * g[c];
  const float sh = bta[c] - mu * sc;
  for (int r = 0; r < 256; ++r)
    out[(size_t)r * 512 + c] = X[(size_t)r * 512 + c] * sc + sh;
}

// ---------------- Kernel 4: single-head logits + log_softmax, one block/batch ----
__global__ void __launch_bounds__(256)
mha2_kernel(const float* __restrict__ K, const float* __restrict__ XN,
            const unsigned char* __restrict__ mask, float* __restrict__ out)
{
  __shared__ float xb[512];
  __shared__ float lg[1024];
  __shared__ float sred[256];

  const int b = blockIdx.x;
  const int tid = threadIdx.x, lane = tid & 31, wave = tid >> 5;
  const int lrow = lane & 15, khalf = lane >> 4;
  const float* Kb = K + (size_t)b * (1024 * 512);
  const unsigned char* mb = mask + (size_t)b * 1024;

  for (int i = tid; i < 128; i += 256)
    ((float4*)xb)[i] = ((const float4*)(XN + (size_t)b * 512))[i];
  __syncthreads();

  for (int t = 0; t < 8; ++t) {
    const int m0 = (t * 8 + wave) << 4;
    v8f acc = {};
    const float* arow = Kb + (size_t)(m0 + lrow) * 512 + 2 * khalf;
    for (int kk = 0; kk < 512; kk += 4) {
      v2f a = nt_v2(arow + kk);
      v2f bq = *(const v2f*)(xb + kk + 2 * khalf);
      v2f bf = (lrow == 0) ? bq : (v2f)(0.f);         // single query column
      acc = wmma4(a, bf, acc);
    }
    if (lrow == 0) {                                  // N==0: lanes 0 and 16
      union { v8f v; float f[8]; } u; u.v = acc;
      #pragma unroll
      for (int r = 0; r < 8; ++r) {
        const int n = m0 + r + 8 * khalf;
        float x = 10.f * tanhf(u.f[r] * 0.044194173824159216f);  // /sqrt(512), clip
        if (mb[n]) x = -3.4028235e38f;
        lg[n] = x;
      }
    }
  }
  __syncthreads();

  // log_softmax over 1024
  float lm = fmaxf(fmaxf(lg[tid], lg[tid + 256]), fmaxf(lg[tid + 512], lg[tid + 768]));
  sred[tid] = lm;
  __syncthreads();
  for (int o = 128; o > 0; o >>= 1) {
    if (tid < o) sred[tid] = fmaxf(sred[tid], sred[tid + o]);
    __syncthreads();
  }
  const float mx = sred[0];
  __syncthreads();
  float ls = expf(lg[tid] - mx) + expf(lg[tid + 256] - mx)
           + expf(lg[tid + 512] - mx) + expf(lg[tid + 768] - mx);
  sred[tid] = ls;
  __syncthreads();
  for (int o = 128; o > 0; o >>= 1) {
    if (tid < o) sred[tid] += sred[tid + o];
    __syncthreads();
  }
  const float lse = logf(sred[0]) + mx;
  __syncthreads();
  #pragma unroll
  for (int i = 0; i < 4; ++i) {
    const int n = tid + i * 256;
    out[(size_t)b * 1024 + n] = lg[n] - lse;
  }
}

extern "C" void kernel_launch(void* const* d_in, const int* in_sizes, int n_in,
                              void* d_out, int out_size, void* d_ws, size_t ws_size,
                              hipStream_t stream) {
  const float* query = (const float*)d_in[0];           // (256,1,512)
  const float* K     = (const float*)d_in[1];           // (256,1024,512)
  const float* V     = (const float*)d_in[2];           // (256,1024,512)
  const unsigned char* mask = (const unsigned char*)d_in[3]; // (256,1,1024) bool
  const float* W1   = (const float*)d_in[4];
  const float* b1   = (const float*)d_in[5];
  const float* W2   = (const float*)d_in[6];
  const float* b2   = (const float*)d_in[7];
  const float* bn_w = (const float*)d_in[8];
  const float* bn_b = (const float*)d_in[9];

  float* ws   = (float*)d_ws;
  float* attn = ws;               // 256*512
  float* h1   = ws + 131072;      // 256*512
  float* q2   = ws + 262144;      // 256*512
  float* xn   = ws + 393216;      // 256*512

  mha1_kernel<<<256, 256, 0, stream>>>(query, K, V, mask, attn);
  ffn_kernel<<<64, 256, 0, stream>>>(attn, W1, b1, nullptr, h1, 1);
  ffn_kernel<<<64, 256, 0, stream>>>(h1, W2, b2, query, q2, 0);
  bn_kernel<<<2, 256, 0, stream>>>(q2, bn_w, bn_b, xn);
  mha2_kernel<<<256, 256, 0, stream>>>(K, xn, mask, (float*)d_out);
}